// BDEqualizer_85109071937758
// MI455X (gfx1250) — compile-verified
//
#include <hip/hip_runtime.h>

// Problem constants (match reference)
#define B_ 16
#define U_ 4
#define A_ 2
#define T_ 8
#define S_ 14
#define F_ 2048

typedef float f32x4 __attribute__((ext_vector_type(4)));

// One block = 256 threads = 8 waves handles 1024 consecutive f-values of one
// (b,u,s) row. Six input streams (h00,h01,h10,h11,y0,y1) are staged into LDS
// with CDNA5 async global->LDS b128 DMA loads (ASYNCcnt), then each lane
// solves four 2x2 systems (one per f in its float4) via Cramer's rule.
__global__ __launch_bounds__(256) void BDEqualizer_kernel(
    const float* __restrict__ y,
    const float* __restrict__ h,
    const int* __restrict__ pind,
    float* __restrict__ out)
{
    // 6 regions x 256 lanes x 16 bytes = 24 KB (<< 320 KB/WGP)
    __shared__ float smem[6 * 256 * 4];

    const int tid    = threadIdx.x;
    const int blk    = blockIdx.x;
    const int fchunk = blk & 1;
    int row = blk >> 1;
    const int s = row % S_;  row /= S_;
    const int u = row % U_;  row /= U_;
    const int b = row;

    const int f0 = fchunk * 1024 + tid * 4;
    const int up = pind[u];            // precoding_ind[0][u] (NTX==1)
    const int t0 = u * A_;             // column block in T

    // y / out layout: (B,U,A,S,F)
    const size_t ybase  = ((((size_t)b * U_ + u) * A_ + 0) * S_ + s) * (size_t)F_ + f0;
    const size_t ystrA  = (size_t)S_ * F_;
    // h layout: (B,U,A,NTX=1,T,S,F)
    const size_t hbase  = (((((size_t)b * U_ + up) * A_ + 0) * T_ + t0) * S_ + s) * (size_t)F_ + f0;
    const size_t hstrA  = (size_t)T_ * S_ * F_;
    const size_t hstrT  = (size_t)S_ * F_;

    const float* gsrc[6];
    gsrc[0] = h + hbase;                  // H[0][0]
    gsrc[1] = h + hbase + hstrT;          // H[0][1]
    gsrc[2] = h + hbase + hstrA;          // H[1][0]
    gsrc[3] = h + hbase + hstrA + hstrT;  // H[1][1]
    gsrc[4] = y + ybase;                  // rhs[0]
    gsrc[5] = y + ybase + ystrA;          // rhs[1]

    // LDS byte offset of this lane's slot (low 32 bits of the flat shared
    // address are the LDS offset on gfx1250).
    const unsigned lds0 = (unsigned)(unsigned long long)(&smem[0]) + (unsigned)tid * 16u;

#pragma unroll
    for (int r = 0; r < 6; ++r) {
        const unsigned lds = lds0 + (unsigned)r * 4096u;
        const unsigned long long ga = (unsigned long long)gsrc[r];
        // CDNA5 async DMA: 16 bytes per lane, global -> LDS, tracked by ASYNCcnt.
        asm volatile("global_load_async_to_lds_b128 %0, %1, off"
                     :: "v"(lds), "v"(ga)
                     : "memory");
    }
    // Wait for this wave's async transfers; each lane reads only its own slots,
    // so no workgroup barrier is required.
    asm volatile("s_wait_asynccnt 0" ::: "memory");

    const f32x4* sm4 = (const f32x4*)smem;
    const f32x4 h00 = sm4[0 * 256 + tid];
    const f32x4 h01 = sm4[1 * 256 + tid];
    const f32x4 h10 = sm4[2 * 256 + tid];
    const f32x4 h11 = sm4[3 * 256 + tid];
    const f32x4 y0  = sm4[4 * 256 + tid];
    const f32x4 y1  = sm4[5 * 256 + tid];

    // Cramer's rule, fp32. Share one reciprocal of det per element (4 IEEE
    // divides instead of 8) -- within fp32 linalg.solve tolerance.
    const f32x4 det  = h00 * h11 - h01 * h10;
    const f32x4 one  = {1.0f, 1.0f, 1.0f, 1.0f};
    const f32x4 rdet = one / det;
    const f32x4 x0   = (y0 * h11 - h01 * y1) * rdet;
    const f32x4 x1   = (h00 * y1 - h10 * y0) * rdet;

    *(f32x4*)(out + ybase)         = x0;   // out[b,u,0,s,f0..f0+3]
    *(f32x4*)(out + ybase + ystrA) = x1;   // out[b,u,1,s,f0..f0+3]
}

extern "C" void kernel_launch(void* const* d_in, const int* in_sizes, int n_in,
                              void* d_out, int out_size, void* d_ws, size_t ws_size,
                              hipStream_t stream) {
    (void)in_sizes; (void)n_in; (void)out_size; (void)d_ws; (void)ws_size;
    const float* y    = (const float*)d_in[0];
    const float* h    = (const float*)d_in[1];
    const int*   pind = (const int*)d_in[2];  // jax default: int64 -> int32
    float* out = (float*)d_out;

    const int blocks = B_ * U_ * S_ * 2;  // 1792
    BDEqualizer_kernel<<<blocks, 256, 0, stream>>>(y, h, pind, out);
}